// MODEL_89404039233977
// MI455X (gfx1250) — compile-verified
//
#include <hip/hip_runtime.h>

// Problem constants (fixed by setup_inputs in the reference)
#define N       4096
#define NS      2048
#define DF      256
#define LAMBDA  0.01f
#define NITER   50
#define NSTEP   8
#define NBIS    26
#define C1ARM   1e-4f

typedef __attribute__((ext_vector_type(2))) float v2f;
typedef __attribute__((ext_vector_type(8))) float v8f;

struct Scal {
  float maxv, sigma;
  float lo0, hi0, lo1, hi1, t0, t1;
  unsigned cnt0, cnt1;
  float w0, w1;
  float thth, obj, fdiv;
  float gTp, pp, thp, tstar;
  float rho, coefA;
};

// ---- block reductions (blockDim.x == 256) ----
__device__ __forceinline__ float bsum(float v, float* sm) {
  int t = threadIdx.x;
  sm[t] = v; __syncthreads();
  for (int s = 128; s > 0; s >>= 1) { if (t < s) sm[t] += sm[t + s]; __syncthreads(); }
  float r = sm[0]; __syncthreads(); return r;
}
__device__ __forceinline__ unsigned busum(unsigned v, unsigned* sm) {
  int t = threadIdx.x;
  sm[t] = v; __syncthreads();
  for (int s = 128; s > 0; s >>= 1) { if (t < s) sm[t] += sm[t + s]; __syncthreads(); }
  unsigned r = sm[0]; __syncthreads(); return r;
}

// ---- stage 0: X = concat(sf, tf), row norms ----
__global__ void k_concat(const float* __restrict__ sf, const float* __restrict__ tf,
                         float* __restrict__ X) {
  size_t i = (size_t)blockIdx.x * blockDim.x + threadIdx.x;
  const size_t half = (size_t)NS * DF;
  if (i < (size_t)N * DF) X[i] = (i < half) ? sf[i] : tf[i - half];
}

__global__ void k_rownorm(const float* __restrict__ X, float* __restrict__ nrm) {
  __shared__ float sm[256];
  float x = X[(size_t)blockIdx.x * DF + threadIdx.x];
  float s = bsum(x * x, sm);
  if (threadIdx.x == 0) nrm[blockIdx.x] = s;
}

// ---- stage 1: Gram matrix G = X X^T via V_WMMA_F32_16X16X4_F32 ----
// One wave per 16x16 output tile; A frag = rows i0..i0+15, B frag = rows j0..j0+15
// (B = X^T block => same per-lane load pattern). 4 waves / 128-thread block.
__global__ void k_gram(const float* __restrict__ X, float* __restrict__ G) {
  const int lane = threadIdx.x & 31;
  const int wave = threadIdx.x >> 5;
  const unsigned tile = blockIdx.x * 4u + wave;     // 65536 tiles total
  const int i0 = (int)(tile >> 8) << 4;
  const int j0 = (int)(tile & 255u) << 4;
  const int m  = lane & 15;
  const int ks = (lane >> 4) << 1;                  // lanes 16-31 hold K=+2,+3
  const float* ar = X + (size_t)(i0 + m) * DF + ks;
  const float* br = X + (size_t)(j0 + m) * DF + ks;
  v8f acc = {0.f, 0.f, 0.f, 0.f, 0.f, 0.f, 0.f, 0.f};
  for (int k = 0; k < DF; k += 4) {
    __builtin_prefetch(ar + k + 64, 0, 0);
    v2f a, b;
    a.x = ar[k]; a.y = ar[k + 1];
    b.x = br[k]; b.y = br[k + 1];
    acc = __builtin_amdgcn_wmma_f32_16x16x4_f32(false, a, false, b, (short)0, acc,
                                                false, false);
  }
  const int rhi = (lane >> 4) << 3;                 // lanes 16-31 hold rows M=8..15
  const int n   = lane & 15;
#pragma unroll
  for (int r = 0; r < 8; ++r)
    G[(size_t)(i0 + rhi + r) * N + j0 + n] = acc[r];
}

// ---- stage 2: sigma = median of off-diag source pd, by rank bisection ----
__global__ void k_init_scal(Scal* sc) {
  if (threadIdx.x == 0) { sc->maxv = 0.f; }
}

__global__ void k_maxpd(const float* __restrict__ G, const float* __restrict__ nrm,
                        Scal* sc) {
  __shared__ float sm[256];
  float m = 0.f;
  const size_t total = (size_t)NS * NS;
  for (size_t idx = (size_t)blockIdx.x * blockDim.x + threadIdx.x; idx < total;
       idx += (size_t)gridDim.x * blockDim.x) {
    size_t i = idx / NS, j = idx - i * NS;
    if (i != j) {
      float pd = nrm[i] + nrm[j] - 2.f * G[i * N + j];
      pd = pd > 0.f ? pd : 0.f;
      m = fmaxf(m, pd);
    }
  }
  sm[threadIdx.x] = m; __syncthreads();
  for (int s = 128; s > 0; s >>= 1) {
    if (threadIdx.x < s) sm[threadIdx.x] = fmaxf(sm[threadIdx.x], sm[threadIdx.x + s]);
    __syncthreads();
  }
  if (threadIdx.x == 0)
    atomicMax((unsigned*)&sc->maxv, __float_as_uint(sm[0]));   // nonneg floats
}

__global__ void k_bisect_init(Scal* sc) {
  if (threadIdx.x == 0) {
    sc->lo0 = sc->lo1 = 0.f;
    sc->hi0 = sc->hi1 = sc->maxv;
    sc->t0 = 0.5f * sc->maxv;
    sc->t1 = 0.5f * sc->maxv;
    sc->cnt0 = 0u; sc->cnt1 = 0u;
  }
}

__global__ void k_count(const float* __restrict__ G, const float* __restrict__ nrm,
                        Scal* sc) {
  __shared__ unsigned sm0[256], sm1[256];
  const float t0 = sc->t0, t1 = sc->t1;
  unsigned c0 = 0, c1 = 0;
  const size_t total = (size_t)NS * NS;
  for (size_t idx = (size_t)blockIdx.x * blockDim.x + threadIdx.x; idx < total;
       idx += (size_t)gridDim.x * blockDim.x) {
    size_t i = idx / NS, j = idx - i * NS;
    if (i != j) {
      float pd = nrm[i] + nrm[j] - 2.f * G[i * N + j];
      pd = pd > 0.f ? pd : 0.f;
      c0 += (pd <= t0);
      c1 += (pd <= t1);
    }
  }
  unsigned r0 = busum(c0, sm0);
  unsigned r1 = busum(c1, sm1);
  if (threadIdx.x == 0) { atomicAdd(&sc->cnt0, r0); atomicAdd(&sc->cnt1, r1); }
}

__global__ void k_bisect_step(Scal* sc) {
  if (threadIdx.x == 0) {
    const unsigned M = (unsigned)NS * (NS - 1);         // even count
    const unsigned k0 = M / 2, k1 = M / 2 + 1;          // two middle ranks
    if (sc->cnt0 >= k0) sc->hi0 = sc->t0; else sc->lo0 = sc->t0;
    if (sc->cnt1 >= k1) sc->hi1 = sc->t1; else sc->lo1 = sc->t1;
    sc->t0 = 0.5f * (sc->lo0 + sc->hi0);
    sc->t1 = 0.5f * (sc->lo1 + sc->hi1);
    sc->cnt0 = 0u; sc->cnt1 = 0u;
  }
}

__global__ void k_set_sigma(Scal* sc) {
  if (threadIdx.x == 0) sc->sigma = 0.5f * (sc->t0 + sc->t1);   // nanmedian (even)
}

// ---- stage 3: K_W built in place over G ----
__global__ void k_buildkw(float* __restrict__ G, const float* __restrict__ nrm,
                          const int* __restrict__ y, const Scal* sc) {
  const float inv_s = 1.f / sc->sigma;
  const size_t total = (size_t)N * N;
  for (size_t idx = (size_t)blockIdx.x * blockDim.x + threadIdx.x; idx < total;
       idx += (size_t)gridDim.x * blockDim.x) {
    size_t i = idx >> 12, j = idx & (N - 1);
    float d = nrm[i] + nrm[j] - 2.f * G[idx];
    G[idx] = (y[i] == y[j]) ? expf(-d * inv_s) : 0.f;
  }
}

__global__ void k_wcount(const int* __restrict__ l, Scal* sc) {
  __shared__ unsigned sm0[256], sm1[256];
  unsigned c0 = 0, c1 = 0;
  for (int i = threadIdx.x; i < N; i += 256) { c0 += (l[i] == 0); c1 += (l[i] == 1); }
  unsigned r0 = busum(c0, sm0);
  unsigned r1 = busum(c1, sm1);
  if (threadIdx.x == 0) { sc->w0 = (float)r0 / N; sc->w1 = (float)r1 / N; }
}

// ---- BFGS machinery ----
__global__ void k_initH(float* __restrict__ H) {
  const size_t total = (size_t)N * N;
  for (size_t idx = (size_t)blockIdx.x * blockDim.x + threadIdx.x; idx < total;
       idx += (size_t)gridDim.x * blockDim.x) {
    size_t i = idx >> 12, j = idx & (N - 1);
    H[idx] = (i == j) ? 1.f : 0.f;
  }
}

__global__ void k_zero(float* __restrict__ v, int n) {
  int i = blockIdx.x * blockDim.x + threadIdx.x;
  if (i < n) v[i] = 0.f;
}

__global__ void k_buildR(const float* __restrict__ v, const int* __restrict__ l,
                         float* __restrict__ R) {
  int j = blockIdx.x * blockDim.x + threadIdx.x;
  if (j < N) {
    int d = l[j];
    R[2 * j + 0] = (d == 0) ? v[j] : 0.f;
    R[2 * j + 1] = (d == 1) ? v[j] : 0.f;
  }
}

// out[i][0:2] = sum_j Kw[i][j] * R[j][0:2]   (Kw row resident in L2, 64 MB/pass)
__global__ void k_matvec2(const float* __restrict__ Kw, const float* __restrict__ R,
                          float* __restrict__ out) {
  __shared__ float sm[256];
  const float* row = Kw + (size_t)blockIdx.x * N;
  float a0 = 0.f, a1 = 0.f;
  for (int j = threadIdx.x; j < N; j += 256) {
    float k = row[j];
    a0 += k * R[2 * j + 0];
    a1 += k * R[2 * j + 1];
  }
  float r0 = bsum(a0, sm);
  float r1 = bsum(a1, sm);
  if (threadIdx.x == 0) { out[2 * blockIdx.x] = r0; out[2 * blockIdx.x + 1] = r1; }
}

// out[i] = scale * sum_j H[i][j] v[j]
__global__ void k_hmatvec(const float* __restrict__ H, const float* __restrict__ v,
                          float* __restrict__ out, float scale) {
  __shared__ float sm[256];
  const float* row = H + (size_t)blockIdx.x * N;
  float a = 0.f;
  for (int j = threadIdx.x; j < N; j += 256) a += row[j] * v[j];
  float r = bsum(a, sm);
  if (threadIdx.x == 0) out[blockIdx.x] = scale * r;
}

// E, f_div, obj, theta.theta at current (A, theta).  Single 256-thread block.
__global__ void k_divpre(const float* __restrict__ A, const float* __restrict__ th,
                         const int* __restrict__ l, float* __restrict__ E, Scal* sc) {
  __shared__ float sm[256];
  const float w0 = sc->w0, w1 = sc->w1;
  float t1 = 0.f, t2 = 0.f, tt = 0.f;
  for (int i = threadIdx.x; i < N; i += 256) {
    float a0 = A[2 * i], a1 = A[2 * i + 1];
    float e0 = expf(a0 - 1.f), e1 = expf(a1 - 1.f);
    int d = l[i];
    t1 += d ? a1 : a0;
    t2 += w0 * e0 + w1 * e1;
    E[2 * i + 0] = (d == 0 ? 1.f : 0.f) - w0 * e0;
    E[2 * i + 1] = (d == 1 ? 1.f : 0.f) - w1 * e1;
    float x = th[i]; tt += x * x;
  }
  float s1 = bsum(t1, sm), s2 = bsum(t2, sm), st = bsum(tt, sm);
  if (threadIdx.x == 0) {
    float fd = (s1 - s2) / N;
    sc->fdiv = fd;
    sc->thth = st;
    sc->obj  = -fd + LAMBDA * st;
  }
}

__global__ void k_grad(const float* __restrict__ GE, const int* __restrict__ l,
                       const float* __restrict__ th, float* __restrict__ g) {
  int j = blockIdx.x * blockDim.x + threadIdx.x;
  if (j < N) g[j] = -GE[2 * j + l[j]] * (1.f / N) + 2.f * LAMBDA * th[j];
}

__global__ void k_gnew_y(const float* __restrict__ GE, const int* __restrict__ l,
                         const float* __restrict__ th, const float* __restrict__ g,
                         float* __restrict__ gn, float* __restrict__ yv) {
  int j = blockIdx.x * blockDim.x + threadIdx.x;
  if (j < N) {
    float v = -GE[2 * j + l[j]] * (1.f / N) + 2.f * LAMBDA * th[j];
    gn[j] = v;
    yv[j] = v - g[j];
  }
}

__global__ void k_dots1(const float* __restrict__ g, const float* __restrict__ p,
                        const float* __restrict__ th, Scal* sc) {
  __shared__ float sm[256];
  float a = 0.f, b = 0.f, c = 0.f;
  for (int i = threadIdx.x; i < N; i += 256) {
    float pi = p[i];
    a += g[i] * pi; b += pi * pi; c += th[i] * pi;
  }
  float ra = bsum(a, sm), rb = bsum(b, sm), rc = bsum(c, sm);
  if (threadIdx.x == 0) { sc->gTp = ra; sc->pp = rb; sc->thp = rc; }
}

// Backtracking line search: f(theta + t p) is elementwise given A and Ap.
__global__ void k_linesearch(const float* __restrict__ A, const float* __restrict__ Ap,
                             const int* __restrict__ l, Scal* sc) {
  __shared__ float sm[256];
  __shared__ float red[2 * NSTEP];
  const float w0 = sc->w0, w1 = sc->w1;
  float tv[NSTEP];
#pragma unroll
  for (int s = 0; s < NSTEP; ++s) tv[s] = 1.f / (float)(1 << s);
  float t1s[NSTEP], t2s[NSTEP];
#pragma unroll
  for (int s = 0; s < NSTEP; ++s) { t1s[s] = 0.f; t2s[s] = 0.f; }
  for (int i = threadIdx.x; i < N; i += 256) {
    float a0 = A[2 * i], a1 = A[2 * i + 1];
    float p0 = Ap[2 * i], p1 = Ap[2 * i + 1];
    int d = l[i];
#pragma unroll
    for (int s = 0; s < NSTEP; ++s) {
      float x0 = a0 + tv[s] * p0, x1 = a1 + tv[s] * p1;
      t1s[s] += d ? x1 : x0;
      t2s[s] += w0 * expf(x0 - 1.f) + w1 * expf(x1 - 1.f);
    }
  }
  for (int s = 0; s < NSTEP; ++s) {
    float r = bsum(t1s[s], sm);
    if (threadIdx.x == 0) red[s] = r;
    r = bsum(t2s[s], sm);
    if (threadIdx.x == 0) red[NSTEP + s] = r;
  }
  __syncthreads();
  if (threadIdx.x == 0) {
    const float obj0 = sc->obj, gTp = sc->gTp, pp = sc->pp, thp = sc->thp,
                thth = sc->thth;
    int   chosen = -1;
    float best_o = 3.0e38f, best_t = tv[NSTEP - 1];
    for (int s = 0; s < NSTEP; ++s) {
      float t = tv[s];
      float fd  = (red[s] - red[NSTEP + s]) / N;
      float reg = thth + 2.f * t * thp + t * t * pp;
      float o   = -fd + LAMBDA * reg;
      if (chosen < 0 && o <= obj0 + C1ARM * t * gTp) chosen = s;  // largest t first
      if (o < best_o) { best_o = o; best_t = t; }
    }
    sc->tstar = (chosen >= 0) ? tv[chosen] : best_t;
  }
}

__global__ void k_update(float* __restrict__ th, float* __restrict__ A,
                         const float* __restrict__ p, const float* __restrict__ Ap,
                         const Scal* sc) {
  int i = blockIdx.x * blockDim.x + threadIdx.x;
  if (i < N) {
    const float t = sc->tstar;
    th[i] += t * p[i];
    A[2 * i + 0] += t * Ap[2 * i + 0];
    A[2 * i + 1] += t * Ap[2 * i + 1];
  }
}

__global__ void k_dots2(const float* __restrict__ p, const float* __restrict__ yv,
                        const float* __restrict__ Hy, Scal* sc) {
  __shared__ float sm[256];
  float a = 0.f, b = 0.f;
  for (int i = threadIdx.x; i < N; i += 256) { a += p[i] * yv[i]; b += yv[i] * Hy[i]; }
  float py = bsum(a, sm), yHy = bsum(b, sm);
  if (threadIdx.x == 0) {
    float sy = sc->tstar * py;
    if (sy > 1e-10f) {
      float rho = 1.f / sy;
      sc->rho = rho;
      sc->coefA = rho * rho * yHy + rho;
    } else {                   // skip curvature update (BFGS safeguard)
      sc->rho = 0.f; sc->coefA = 0.f;
    }
  }
}

// H += -rho*(s_i Hy_j + Hy_i s_j) + coefA * s_i * s_j,   s = tstar * p
__global__ void k_hupdate(float* __restrict__ H, const float* __restrict__ p,
                          const float* __restrict__ Hy, const Scal* sc) {
  const float t = sc->tstar, rho = sc->rho, cA = sc->coefA;
  const size_t total = (size_t)N * N;
  for (size_t idx = (size_t)blockIdx.x * blockDim.x + threadIdx.x; idx < total;
       idx += (size_t)gridDim.x * blockDim.x) {
    size_t i = idx >> 12, j = idx & (N - 1);
    float si = t * p[i], sj = t * p[j];
    H[idx] += -rho * (si * Hy[j] + Hy[i] * sj) + cA * si * sj;
  }
}

__global__ void k_copy(float* __restrict__ dst, const float* __restrict__ src, int n) {
  int i = blockIdx.x * blockDim.x + threadIdx.x;
  if (i < n) dst[i] = src[i];
}

__global__ void k_writeout(const Scal* sc, float* __restrict__ out) {
  if (threadIdx.x == 0) out[0] = sc->fdiv;
}

extern "C" void kernel_launch(void* const* d_in, const int* in_sizes, int n_in,
                              void* d_out, int out_size, void* d_ws, size_t ws_size,
                              hipStream_t stream) {
  (void)in_sizes; (void)n_in; (void)out_size; (void)ws_size;
  const float* sf = (const float*)d_in[0];
  const float* tf = (const float*)d_in[1];
  const int*   yy = (const int*)d_in[2];
  const int*   ll = (const int*)d_in[3];
  float* out = (float*)d_out;

  // Workspace layout (~139 MB): G/K_W and H both live in the 192 MB L2.
  float* ws  = (float*)d_ws;
  float* G   = ws;                          // N*N   (becomes K_W in place)
  float* H   = G + (size_t)N * N;           // N*N
  float* X   = H + (size_t)N * N;           // N*DF
  float* nrm = X + (size_t)N * DF;          // N
  float* th  = nrm + N;                     // N
  float* g   = th + N;                      // N
  float* gn  = g + N;                       // N
  float* p   = gn + N;                      // N
  float* yv  = p + N;                       // N
  float* Hy  = yv + N;                      // N
  float* A   = Hy + N;                      // 2N
  float* Ap  = A + 2 * N;                   // 2N
  float* E   = Ap + 2 * N;                  // 2N
  float* GE  = E + 2 * N;                   // 2N
  float* R   = GE + 2 * N;                  // 2N
  Scal*  sc  = (Scal*)(R + 2 * N);

  const dim3 b256(256);
  const dim3 gN((N + 255) / 256);           // 16 blocks for N-length elementwise
  const dim3 gNDF((N * DF + 255) / 256);
  const dim3 gNN(8192);                     // grid-stride over N*N
  const dim3 gNS(4096);                     // grid-stride over NS*NS

  // Stage 0/1: X, norms, Gram via f32 WMMA
  k_concat<<<gNDF, b256, 0, stream>>>(sf, tf, X);
  k_rownorm<<<dim3(N), b256, 0, stream>>>(X, nrm);
  k_gram<<<dim3((N / 16) * (N / 16) / 4), dim3(128), 0, stream>>>(X, G);

  // Stage 2: sigma by rank bisection on source block
  k_init_scal<<<dim3(1), dim3(64), 0, stream>>>(sc);
  k_maxpd<<<gNS, b256, 0, stream>>>(G, nrm, sc);
  k_bisect_init<<<dim3(1), dim3(64), 0, stream>>>(sc);
  for (int it = 0; it < NBIS; ++it) {
    k_count<<<gNS, b256, 0, stream>>>(G, nrm, sc);
    k_bisect_step<<<dim3(1), dim3(64), 0, stream>>>(sc);
  }
  k_set_sigma<<<dim3(1), dim3(64), 0, stream>>>(sc);

  // Stage 3: K_W in place, domain weights, BFGS init
  k_buildkw<<<gNN, b256, 0, stream>>>(G, nrm, yy, sc);
  k_wcount<<<dim3(1), b256, 0, stream>>>(ll, sc);
  k_initH<<<gNN, b256, 0, stream>>>(H);
  k_zero<<<gN, b256, 0, stream>>>(th, N);

  // initial f, grad at theta = 0
  k_buildR<<<gN, b256, 0, stream>>>(th, ll, R);
  k_matvec2<<<dim3(N), b256, 0, stream>>>(G, R, A);
  k_divpre<<<dim3(1), b256, 0, stream>>>(A, th, ll, E, sc);
  k_matvec2<<<dim3(N), b256, 0, stream>>>(G, E, GE);
  k_grad<<<gN, b256, 0, stream>>>(GE, ll, th, g);

  // Stage 4: 50 fixed BFGS iterations (all control flow on-device via sc)
  for (int it = 0; it < NITER; ++it) {
    k_hmatvec<<<dim3(N), b256, 0, stream>>>(H, g, p, -1.f);       // p = -H g
    k_dots1<<<dim3(1), b256, 0, stream>>>(g, p, th, sc);
    k_buildR<<<gN, b256, 0, stream>>>(p, ll, R);
    k_matvec2<<<dim3(N), b256, 0, stream>>>(G, R, Ap);            // Ap = K_W (p.oh)
    k_linesearch<<<dim3(1), b256, 0, stream>>>(A, Ap, ll, sc);    // picks tstar
    k_update<<<gN, b256, 0, stream>>>(th, A, p, Ap, sc);          // theta, A += t*
    k_divpre<<<dim3(1), b256, 0, stream>>>(A, th, ll, E, sc);     // new f, E, obj
    k_matvec2<<<dim3(N), b256, 0, stream>>>(G, E, GE);
    k_gnew_y<<<gN, b256, 0, stream>>>(GE, ll, th, g, gn, yv);     // g_new, y
    k_hmatvec<<<dim3(N), b256, 0, stream>>>(H, yv, Hy, 1.f);      // Hy
    k_dots2<<<dim3(1), b256, 0, stream>>>(p, yv, Hy, sc);         // rho, coefA
    k_hupdate<<<gNN, b256, 0, stream>>>(H, p, Hy, sc);            // rank-2 update
    k_copy<<<gN, b256, 0, stream>>>(g, gn, N);
  }

  // Output: divergence at theta_hat (last k_divpre wrote it)
  k_writeout<<<dim3(1), dim3(64), 0, stream>>>(sc, out);
}